// HGTLayer_35682588295606
// MI455X (gfx1250) — compile-verified
//
#include <hip/hip_runtime.h>
#include <math.h>

#define DIM 128
#define NH  4
#define DK  32

typedef float v2f __attribute__((ext_vector_type(2)));
typedef float v8f __attribute__((ext_vector_type(8)));

// ---------------------------------------------------------------------------
// CDNA5 fp32 WMMA: D(16x16,f32) = A(16x4,f32) x B(4x16,f32) + C
// A lane layout: lanes0-15 -> {K=0,K=1}, lanes16-31 -> {K=2,K=3}, M = lane&15
// B lane layout: lanes0-15 -> {K=0,K=1}, lanes16-31 -> {K=2,K=3}, N = lane&15
// C/D: VGPR j, lane L -> M = j + 8*(L>>4), N = L&15
// ---------------------------------------------------------------------------
__device__ __forceinline__ v8f wmma_f32_16x16x4(v2f a, v2f b, v8f c) {
#if __has_builtin(__builtin_amdgcn_wmma_f32_16x16x4_f32)
  return __builtin_amdgcn_wmma_f32_16x16x4_f32(false, a, false, b, (short)0, c,
                                               false, false);
#else
  // safety fallback so compilation never hard-fails (not numerically correct)
  c[0] += a.x * b.x + a.y * b.y;
  return c;
#endif
}

// Fold per-head relation matrices into the 128x128 projection weights:
// Wk_eff[r][p][h*32+j] = sum_i Wk[src(r)][p][h*32+i] * rel_att[r][h][i][j]
__global__ void fold_rel_weights(const float* __restrict__ Wk,
                                 const float* __restrict__ bk,
                                 const float* __restrict__ Wv,
                                 const float* __restrict__ bv,
                                 const float* __restrict__ rel_att,
                                 const float* __restrict__ rel_msg,
                                 float* __restrict__ WkE, float* __restrict__ bkE,
                                 float* __restrict__ WvE, float* __restrict__ bvE)
{
  const int r   = blockIdx.y;                              // relation 0..2
  const int idx = blockIdx.x * blockDim.x + threadIdx.x;   // 0..16383
  const int p   = idx >> 7;          // input row
  const int c   = idx & (DIM - 1);   // output column
  const int h   = c >> 5;
  const int j   = c & (DK - 1);
  const int s   = (r == 0) ? 0 : 1;  // RELS src types: (0,_),(1,_),(1,_)

  const float* A     = rel_att + (size_t)(r * NH + h) * DK * DK;
  const float* M     = rel_msg + (size_t)(r * NH + h) * DK * DK;
  const float* wkRow = Wk + (size_t)s * DIM * DIM + (size_t)p * DIM + h * DK;
  const float* wvRow = Wv + (size_t)s * DIM * DIM + (size_t)p * DIM + h * DK;

  float aK = 0.f, aV = 0.f;
  #pragma unroll 8
  for (int i = 0; i < DK; ++i) {
    aK += wkRow[i] * A[i * DK + j];
    aV += wvRow[i] * M[i * DK + j];
  }
  WkE[(size_t)r * DIM * DIM + idx] = aK;
  WvE[(size_t)r * DIM * DIM + idx] = aV;

  if (p == 0) {
    const float* bkS = bk + s * DIM + h * DK;
    const float* bvS = bv + s * DIM + h * DK;
    float bK = 0.f, bV = 0.f;
    #pragma unroll 8
    for (int i = 0; i < DK; ++i) {
      bK += bkS[i] * A[i * DK + j];
      bV += bvS[i] * M[i * DK + j];
    }
    bkE[r * DIM + c] = bK;
    bvE[r * DIM + c] = bV;
  }
}

// Unified 128-wide WMMA GEMM:
//   C = (X*ascale) @ W + bias                       (hin == nullptr)
//   C = alpha*((X*ascale)@W + bias) + (1-alpha)*hin (hin != nullptr,
//                                                    alpha = sigmoid(*skipPtr))
// W staged once per block into LDS (exactly 64 KB), K-row-pair interleaved:
//   element(k,c) -> (k>>1)*256 + 2c + (k&1) + ((k>>1)&1 ? 0 : 32)
// So each B fragment {W[ka][col], W[ka+1][col]} is ONE aligned ds_load_b64
// landing directly in the WMMA operand pair. The 32-float skew makes the two
// half-waves of every load hit disjoint 32-bank halves (conflict-free), and
// it depends only on kSel, so the K-loop is one base pointer + literal
// offsets: no per-load VALU, no register shuffles.
__global__ __launch_bounds__(256) void gemm128_wmma(
    const float* __restrict__ X, const float* __restrict__ W,
    const float* __restrict__ bias, float* __restrict__ C,
    const float* __restrict__ hin, const float* __restrict__ skipPtr,
    int nrows, float ascale)
{
  __shared__ float wt[DIM * DIM];  // exactly 64 KB (max idx = 16383)

  // cooperative stage: two coalesced float4 row reads -> interleaved b64 pairs
  for (int j = threadIdx.x; j < (DIM * DIM) / 8; j += 256) {
    const int p  = j >> 5;          // K-row pair 0..63
    const int cq = (j & 31) << 2;   // column quad 0,4,...,124
    const float4 r0 = *(const float4*)(W + (size_t)(2 * p) * DIM + cq);
    const float4 r1 = *(const float4*)(W + (size_t)(2 * p + 1) * DIM + cq);
    float* dp = wt + p * 256 + cq * 2 + ((p & 1) ? 0 : 32);
    ((v2f*)dp)[0] = (v2f){r0.x, r1.x};
    ((v2f*)dp)[1] = (v2f){r0.y, r1.y};
    ((v2f*)dp)[2] = (v2f){r0.z, r1.z};
    ((v2f*)dp)[3] = (v2f){r0.w, r1.w};
  }
  __syncthreads();

  const int lane      = threadIdx.x & 31;
  const int wave      = threadIdx.x >> 5;
  const int nRowTiles = (nrows + 15) >> 4;
  const int rowTile   = blockIdx.x * 8 + wave;
  if (rowTile >= nRowTiles) return;           // whole-wave exit keeps EXEC all-1s
  const int row0  = rowTile << 4;
  int mrow        = row0 + (lane & 15);
  if (mrow >= nrows) mrow = nrows - 1;        // clamp loads on tail tile
  const int kSel  = (lane >> 4) << 1;         // 0 or 2
  const int colLo = lane & 15;

  v8f acc[8];
  #pragma unroll
  for (int ct = 0; ct < 8; ++ct) {
    const float b0 = bias[ct * 16 + colLo];
    #pragma unroll
    for (int j = 0; j < 8; ++j) acc[ct][j] = b0;
  }

  // per-lane invariant LDS base: row-pair (kSel>>1), skew(kSel=0)=32, skew(2)=0
  const float* __restrict__ wrow = wt + colLo * 2 + (kSel ? 256 : 32);
  const float* __restrict__ xr   = X + (size_t)mrow * DIM;

  for (int k0 = 0; k0 < DIM; k0 += 4) {
    const int ka = k0 + kSel;
    v2f a;
    a.x = xr[ka] * ascale;
    a.y = xr[ka + 1] * ascale;
    #pragma unroll
    for (int ct = 0; ct < 8; ++ct) {
      const v2f b = *(const v2f*)(wrow + k0 * 128 + ct * 32);  // one ds_load_b64
      acc[ct] = wmma_f32_16x16x4(a, b, acc[ct]);
    }
  }

  float alpha = 1.0f, beta = 0.0f;
  if (skipPtr) {
    alpha = 1.0f / (1.0f + __expf(-skipPtr[0]));
    beta  = 1.0f - alpha;
  }

  const int rbase = row0 + ((lane >> 4) << 3);
  #pragma unroll
  for (int ct = 0; ct < 8; ++ct) {
    #pragma unroll
    for (int j = 0; j < 8; ++j) {
      const int rr = rbase + j;
      if (rr < nrows) {
        const size_t off = (size_t)rr * DIM + ct * 16 + colLo;
        C[off] = hin ? (acc[ct][j] * alpha + hin[off] * beta) : acc[ct][j];
      }
    }
  }
}

// Sign-split float atomic max (works for mixed signs; init bytes = 0xFF).
__device__ __forceinline__ void atomicMaxF(float* addr, float v) {
  if (v >= 0.f) atomicMax((int*)addr, __float_as_int(v));
  else          atomicMin((unsigned int*)addr, __float_as_uint(v));
}

// score[e,h] = dot(Q[dst], K[src]) * pri[h] / sqrt(DK); segment-max into mbuf
__global__ void edge_score(const float* __restrict__ Q, const float* __restrict__ K,
                           const int* __restrict__ src, const int* __restrict__ dst,
                           const float* __restrict__ pri, float* __restrict__ score,
                           float* __restrict__ mbuf, int E)
{
  const int idx = blockIdx.x * blockDim.x + threadIdx.x;
  if (idx >= E * NH) return;
  const int e = idx >> 2;
  const int h = idx & 3;
  const int s = src[e];
  const int d = dst[e];
  const float4* q = (const float4*)(Q + (size_t)d * DIM + h * DK);
  const float4* k = (const float4*)(K + (size_t)s * DIM + h * DK);
  float acc = 0.f;
  #pragma unroll
  for (int i = 0; i < DK / 4; ++i) {
    const float4 qa = q[i], ka = k[i];
    acc += qa.x * ka.x + qa.y * ka.y + qa.z * ka.z + qa.w * ka.w;
  }
  const float scv = acc * pri[h] * 0.17677669529663688f;  // 1/sqrt(32)
  score[idx] = scv;
  atomicMaxF(&mbuf[(size_t)d * NH + h], scv);
}

// score -> exp(score - m[dst]); segment-sum into zbuf
__global__ void edge_expsum(float* __restrict__ score, const int* __restrict__ dst,
                            const float* __restrict__ mbuf, float* __restrict__ zbuf,
                            int E)
{
  const int idx = blockIdx.x * blockDim.x + threadIdx.x;
  if (idx >= E * NH) return;
  const int e = idx >> 2;
  const int h = idx & 3;
  const int d = dst[e];
  const float ex = __expf(score[idx] - mbuf[(size_t)d * NH + h]);
  score[idx] = ex;
  atomicAdd(&zbuf[(size_t)d * NH + h], ex);
}

// agg[dst] += V[src] * (e / z[dst]); 32 lanes per edge, float4 per lane
__global__ void edge_message(const float* __restrict__ V, const int* __restrict__ src,
                             const int* __restrict__ dst, const float* __restrict__ escore,
                             const float* __restrict__ zbuf, float* __restrict__ agg,
                             int E)
{
  const int idx = blockIdx.x * blockDim.x + threadIdx.x;
  const int e   = idx >> 5;
  if (e >= E) return;
  const int lane = idx & 31;
  const int h    = lane >> 3;           // 8 lanes (32 floats) per head
  const int s = src[e];
  const int d = dst[e];
  const float a = escore[(size_t)e * NH + h] / zbuf[(size_t)d * NH + h];
  const float4 v = *(const float4*)(V + (size_t)s * DIM + lane * 4);
  float* op = agg + (size_t)d * DIM + lane * 4;
  atomicAdd(op + 0, v.x * a);
  atomicAdd(op + 1, v.y * a);
  atomicAdd(op + 2, v.z * a);
  atomicAdd(op + 3, v.w * a);
}

extern "C" void kernel_launch(void* const* d_in, const int* in_sizes, int n_in,
                              void* d_out, int out_size, void* d_ws, size_t ws_size,
                              hipStream_t stream)
{
  (void)n_in; (void)out_size; (void)ws_size;

  const float* h0      = (const float*)d_in[0];
  const float* h1      = (const float*)d_in[1];
  const float* Wk      = (const float*)d_in[2];
  const float* bk      = (const float*)d_in[3];
  const float* Wq      = (const float*)d_in[4];
  const float* bq      = (const float*)d_in[5];
  const float* Wv      = (const float*)d_in[6];
  const float* bv      = (const float*)d_in[7];
  const float* Wa      = (const float*)d_in[8];
  const float* ba      = (const float*)d_in[9];
  const float* rel_att = (const float*)d_in[10];
  const float* rel_msg = (const float*)d_in[11];
  const float* rel_pri = (const float*)d_in[12];
  const float* skip    = (const float*)d_in[13];
  const int* srcs[3] = {(const int*)d_in[14], (const int*)d_in[16], (const int*)d_in[18]};
  const int* dsts[3] = {(const int*)d_in[15], (const int*)d_in[17], (const int*)d_in[19]};
  const int  Ecnt[3] = {in_sizes[14], in_sizes[16], in_sizes[18]};

  const int N = in_sizes[0] / DIM;
  const float* hs[2]  = {h0, h1};
  const int relSrc[3] = {0, 1, 1};   // RELS = [(0,1),(1,0),(1,1)]
  const int relDst[3] = {1, 0, 1};

  // ---- workspace carve-out (floats) ----
  float* ws = (float*)d_ws;
  size_t o = 0;
  float* WkE = ws + o; o += (size_t)3 * DIM * DIM;
  float* WvE = ws + o; o += (size_t)3 * DIM * DIM;
  float* bkE = ws + o; o += 3 * DIM;
  float* bvE = ws + o; o += 3 * DIM;
  float* Kr  = ws + o; o += (size_t)3 * N * DIM;
  float* Vr  = ws + o; o += (size_t)3 * N * DIM;
  float* Qt  = ws + o; o += (size_t)2 * N * DIM;
  float* mb  = ws + o; o += (size_t)3 * N * NH;
  float* zb  = ws + o; o += (size_t)3 * N * NH;
  float* agg = ws + o; o += (size_t)2 * N * DIM;
  float* sc  = ws + o;   // (E0+E1+E2) * NH

  // ---- init (graph-capture-safe async memsets) ----
  hipMemsetAsync(mb, 0xFF, (size_t)3 * N * NH * sizeof(float), stream);  // -NaN sorts below all
  hipMemsetAsync(zb, 0x00, (size_t)3 * N * NH * sizeof(float), stream);
  hipMemsetAsync(agg, 0x00, (size_t)2 * N * DIM * sizeof(float), stream);

  // ---- fold relation matrices into projection weights ----
  fold_rel_weights<<<dim3(DIM * DIM / 256, 3), 256, 0, stream>>>(
      Wk, bk, Wv, bv, rel_att, rel_msg, WkE, bkE, WvE, bvE);

  const int nRowTiles  = (N + 15) / 16;
  const unsigned gemmBlocks = (unsigned)((nRowTiles + 7) / 8);

  // Q projections (per destination type)
  for (int t = 0; t < 2; ++t)
    gemm128_wmma<<<gemmBlocks, 256, 0, stream>>>(
        hs[t], Wq + (size_t)t * DIM * DIM, bq + t * DIM,
        Qt + (size_t)t * N * DIM, nullptr, nullptr, N, 1.0f);

  // per-relation fused K,V projections
  for (int r = 0; r < 3; ++r) {
    const float* x = hs[relSrc[r]];
    gemm128_wmma<<<gemmBlocks, 256, 0, stream>>>(
        x, WkE + (size_t)r * DIM * DIM, bkE + r * DIM,
        Kr + (size_t)r * N * DIM, nullptr, nullptr, N, 1.0f);
    gemm128_wmma<<<gemmBlocks, 256, 0, stream>>>(
        x, WvE + (size_t)r * DIM * DIM, bvE + r * DIM,
        Vr + (size_t)r * N * DIM, nullptr, nullptr, N, 1.0f);
  }

  // ---- edge phases per relation ----
  size_t scOff = 0;
  for (int r = 0; r < 3; ++r) {
    const int E  = Ecnt[r];
    float* score = sc + scOff;  scOff += (size_t)E * NH;
    const float* Q = Qt + (size_t)relDst[r] * N * DIM;
    const float* K = Kr + (size_t)r * N * DIM;
    float* m_r = mb + (size_t)r * N * NH;
    float* z_r = zb + (size_t)r * N * NH;
    const unsigned bScore = (unsigned)(((size_t)E * NH + 255) / 256);
    const unsigned bMsg   = (unsigned)(((size_t)E * 32 + 255) / 256);

    edge_score<<<bScore, 256, 0, stream>>>(Q, K, srcs[r], dsts[r],
                                           rel_pri + r * NH, score, m_r, E);
    edge_expsum<<<bScore, 256, 0, stream>>>(score, dsts[r], m_r, z_r, E);
    edge_message<<<bMsg, 256, 0, stream>>>(Vr + (size_t)r * N * DIM, srcs[r],
                                           dsts[r], score, z_r,
                                           agg + (size_t)relDst[r] * N * DIM, E);
  }

  // ---- output projection + skip blend (mean folded into A-scale) ----
  float* out = (float*)d_out;
  const float invcnt[2] = {1.0f, 0.5f};   // agg[0] from 1 relation, agg[1] from 2
  for (int t = 0; t < 2; ++t)
    gemm128_wmma<<<gemmBlocks, 256, 0, stream>>>(
        agg + (size_t)t * N * DIM, Wa + (size_t)t * DIM * DIM, ba + t * DIM,
        out + (size_t)t * N * DIM, hs[t], skip + t, N, invcnt[t]);
}